// ConvLayer_88897233092924
// MI455X (gfx1250) — compile-verified
//
#include <hip/hip_runtime.h>
#include <cmath>

#define HH 32
#define WW 32
#define CC 3
#define PHH 5
#define PWW 5
#define HP 28
#define WP 28
#define PPATCH (HP*WP)        /* 784 */
#define NN 32
#define XX (PPATCH*NN)        /* 25088 = 196*128 */
#define LL 75
#define LPAD 80               /* K multiple of 16 */
#define MM 384                /* 6*64 */
#define GG 2
#define JITTER 1e-6f
#define MTILES (MM/64)        /* 6 */

#define AS_PAD 20             /* A-tile row stride (floats): 16B-aligned rows, distinct banks */
#define BS_PAD 132            /* B-tile row stride (floats): 528B rows, 16B aligned */
#define AS_STRIDE (64*AS_PAD) /* 1280 */
#define BS_STRIDE (16*BS_PAD) /* 2112 */

typedef __attribute__((ext_vector_type(2))) float v2f;
typedef __attribute__((ext_vector_type(8))) float v8f;

// ---------------- patch extraction: PatT[l][x], x = p*NN + n ----------------
__global__ __launch_bounds__(256) void patches_kernel(const float* __restrict__ ndx,
                                                      float* __restrict__ patT) {
  int idx = blockIdx.x * 256 + threadIdx.x;
  if (idx >= LPAD * XX) return;
  int l = idx / XX, x = idx - l * XX;
  float v = 0.f;
  if (l < LL) {
    int p = x / NN, n = x - p * NN;
    int ph = p / WP, pw = p - ph * WP;
    int d = l / CC, c = l - d * CC;
    int di = d / PWW, dj = d - di * PWW;
    v = ndx[n * (HH * WW * CC) + ((ph + di) * WW + (pw + dj)) * CC + c];
  }
  patT[l * XX + x] = v;
}

__global__ __launch_bounds__(256) void pn_kernel(const float* __restrict__ patT,
                                                 float* __restrict__ pn) {
  int x = blockIdx.x * 256 + threadIdx.x;
  if (x >= XX) return;
  float s = 0.f;
  for (int l = 0; l < LL; ++l) { float v = patT[l * XX + x]; s += v * v; }
  pn[x] = s;
}

// ---------------- small dense LA (per g) ----------------
__global__ void zn_kernel(const float* __restrict__ Zg, float* __restrict__ zn) {
  int m = blockIdx.x * 256 + threadIdx.x;
  if (m >= MM) return;
  float s = 0.f;
  for (int l = 0; l < LL; ++l) { float v = Zg[m * LL + l]; s += v * v; }
  zn[m] = s;
}

__global__ __launch_bounds__(256) void zp_kernel(const float* __restrict__ Zg,
                                                 float* __restrict__ Zp) {
  int idx = blockIdx.x * 256 + threadIdx.x;
  if (idx >= MM * LPAD) return;
  int m = idx / LPAD, k = idx - m * LPAD;
  Zp[idx] = (k < LL) ? Zg[m * LL + k] : 0.f;
}

__global__ __launch_bounds__(256) void kuu_kernel(const float* __restrict__ Zg,
                                                  const float* __restrict__ zn,
                                                  const float* __restrict__ varp,
                                                  const float* __restrict__ lsp,
                                                  float* __restrict__ Kuu) {
  int idx = blockIdx.x * 256 + threadIdx.x;
  if (idx >= MM * MM) return;
  int i = idx / MM, j = idx - i * MM;
  float dot = 0.f;
  for (int l = 0; l < LL; ++l) dot += Zg[i * LL + l] * Zg[j * LL + l];
  float ls = lsp[0];
  float v = varp[0] * __expf(-0.5f * (zn[i] + zn[j] - 2.f * dot) / (ls * ls));
  if (i == j) v += JITTER;
  Kuu[idx] = v;
}

// in-place right-looking Cholesky (lower), single block
__global__ __launch_bounds__(256) void chol_kernel(float* __restrict__ K) {
  int tid = threadIdx.x;
  for (int j = 0; j < MM; ++j) {
    __syncthreads();
    if (tid == 0) K[j * MM + j] = sqrtf(K[j * MM + j]);
    __syncthreads();
    float d = K[j * MM + j];
    for (int i = j + 1 + tid; i < MM; i += 256) K[i * MM + j] /= d;
    __syncthreads();
    for (int i = j + 1 + tid; i < MM; i += 256) {
      float lij = K[i * MM + j];
      for (int k = j + 1; k <= i; ++k) K[i * MM + k] -= lij * K[k * MM + j];
    }
  }
}

// W = L^-1 (lower triangular inverse), one thread per column
__global__ void winv_kernel(const float* __restrict__ L, float* __restrict__ W) {
  int j = blockIdx.x * 256 + threadIdx.x;
  if (j >= MM) return;
  W[j * MM + j] = 1.f / L[j * MM + j];
  for (int i = j + 1; i < MM; ++i) {
    float s = 0.f;
    for (int k = j; k < i; ++k) s += L[i * MM + k] * W[k * MM + j];
    W[i * MM + j] = -s / L[i * MM + i];
  }
}

// S = W^T W  (Kuu^-1)
__global__ __launch_bounds__(256) void s_kernel(const float* __restrict__ W,
                                                float* __restrict__ S) {
  int idx = blockIdx.x * 256 + threadIdx.x;
  if (idx >= MM * MM) return;
  int i = idx / MM, j = idx - i * MM;
  int k0 = i > j ? i : j;
  float s = 0.f;
  for (int k = k0; k < MM; ++k) s += W[k * MM + i] * W[k * MM + j];
  S[idx] = s;
}

// LqT[k][m] = tril(q_sqrt[g])[m][k]
__global__ __launch_bounds__(256) void lqt_kernel(const float* __restrict__ qs, int g,
                                                  float* __restrict__ LqT) {
  int idx = blockIdx.x * 256 + threadIdx.x;
  if (idx >= MM * MM) return;
  int k = idx / MM, m = idx - k * MM;
  LqT[idx] = (m >= k) ? qs[(size_t)g * MM * MM + m * MM + k] : 0.f;
}

// ---------------- WMMA f32 GEMM: D(MxXX) = A(MxK) * B(KxXX), tiles 64x128 ----------------
// Double-buffered LDS, async direct-to-LDS staging (global_load_async_to_lds_b128),
// f32 WMMA 16x16x4 compute.
// MODE 0: store D.  MODE 1: RBF epilogue -> D.  MODE 2: per-column sum of squares partials.
template <int MODE>
__global__ __launch_bounds__(256) void gemm_wmma(const float* __restrict__ A,
                                                 const float* __restrict__ B,
                                                 float* __restrict__ D, int K,
                                                 const float* __restrict__ zn,
                                                 const float* __restrict__ pnv,
                                                 const float* __restrict__ varp,
                                                 const float* __restrict__ lsp) {
  __shared__ float As[2 * AS_STRIDE];
  __shared__ float Bs[2 * BS_STRIDE];
  __shared__ float red[8 * 64];
  const int tid = threadIdx.x;
  const int wave = tid >> 5, lane = tid & 31;
  const int waveM = wave >> 2, waveX = wave & 3;
  const int half = lane >> 4, l15 = lane & 15;
  const int mBase0 = blockIdx.y * 64;
  const int xBase0 = blockIdx.x * 128;

  v8f acc00 = {}, acc01 = {}, acc10 = {}, acc11 = {};
  const int ktiles = K >> 4;

  // Issue one K-tile's worth of async global->LDS copies into buffer `buf`.
  auto stage = [&](int kt, int buf) {
    {
      // A tile 64x16: one b128 per thread
      int e = tid << 2;
      int r = e >> 4, c = e & 15;
      unsigned lds = (unsigned)(uintptr_t)&As[buf * AS_STRIDE + r * AS_PAD + c];
      unsigned goff = (unsigned)((((mBase0 + r) * K + (kt << 4) + c)) << 2);
      asm volatile("global_load_async_to_lds_b128 %0, %1, %2 offset:0"
                   :: "v"(lds), "v"(goff), "s"(A) : "memory");
    }
#pragma unroll
    for (int q = 0; q < 2; ++q) {
      // B tile 16x128: two b128 per thread
      int e = (tid << 2) + (q << 10);
      int r = e >> 7, c = e & 127;
      unsigned lds = (unsigned)(uintptr_t)&Bs[buf * BS_STRIDE + r * BS_PAD + c];
      unsigned goff = (unsigned)(((((kt << 4) + r) * XX) + xBase0 + c) << 2);
      asm volatile("global_load_async_to_lds_b128 %0, %1, %2 offset:0"
                   :: "v"(lds), "v"(goff), "s"(B) : "memory");
    }
  };

  stage(0, 0);
  asm volatile("s_wait_asynccnt 0x0" ::: "memory");
  __syncthreads();

  for (int kt = 0; kt < ktiles; ++kt) {
    const int buf = kt & 1;
    if (kt + 1 < ktiles) stage(kt + 1, buf ^ 1);  // overlap DMA with WMMA

    const float* as = &As[buf * AS_STRIDE];
    const float* bs = &Bs[buf * BS_STRIDE];
#pragma unroll
    for (int kk = 0; kk < 16; kk += 4) {
      int kb = kk + 2 * half;
      v2f a0 = *(const v2f*)&as[(waveM * 32 + l15) * AS_PAD + kb];
      v2f a1 = *(const v2f*)&as[(waveM * 32 + 16 + l15) * AS_PAD + kb];
      v2f b0, b1;
      b0.x = bs[kb * BS_PAD + waveX * 32 + l15];
      b0.y = bs[(kb + 1) * BS_PAD + waveX * 32 + l15];
      b1.x = bs[kb * BS_PAD + waveX * 32 + 16 + l15];
      b1.y = bs[(kb + 1) * BS_PAD + waveX * 32 + 16 + l15];
      acc00 = __builtin_amdgcn_wmma_f32_16x16x4_f32(false, a0, false, b0, (short)0, acc00, false, false);
      acc01 = __builtin_amdgcn_wmma_f32_16x16x4_f32(false, a0, false, b1, (short)0, acc01, false, false);
      acc10 = __builtin_amdgcn_wmma_f32_16x16x4_f32(false, a1, false, b0, (short)0, acc10, false, false);
      acc11 = __builtin_amdgcn_wmma_f32_16x16x4_f32(false, a1, false, b1, (short)0, acc11, false, false);
    }
    asm volatile("s_wait_asynccnt 0x0" ::: "memory");
    __syncthreads();
  }

  if (MODE == 2) {
    float s0 = 0.f, s1 = 0.f;
#pragma unroll
    for (int r = 0; r < 8; ++r) {
      s0 += acc00[r] * acc00[r] + acc10[r] * acc10[r];
      s1 += acc01[r] * acc01[r] + acc11[r] * acc11[r];
    }
    red[wave * 64 + lane] = s0;
    red[wave * 64 + 32 + lane] = s1;
    __syncthreads();
    if (tid < 128) {
      int c = tid;
      int wx = c >> 5, j = (c >> 4) & 1, lo = c & 15;
      float s = 0.f;
#pragma unroll
      for (int wm = 0; wm < 2; ++wm) {
        int w = wm * 4 + wx;
        s += red[w * 64 + j * 32 + lo] + red[w * 64 + j * 32 + 16 + lo];
      }
      D[blockIdx.y * XX + xBase0 + c] = s;  // sqpart[mtile][x]
    }
  } else {
    float variance = (MODE == 1) ? varp[0] : 0.f;
    float sc = (MODE == 1) ? (-0.5f / (lsp[0] * lsp[0])) : 0.f;
    const v8f* accs[2][2] = {{&acc00, &acc01}, {&acc10, &acc11}};
#pragma unroll
    for (int i = 0; i < 2; ++i)
#pragma unroll
      for (int j = 0; j < 2; ++j) {
        int m = mBase0 + waveM * 32 + i * 16 + 8 * half;
        int x = xBase0 + waveX * 32 + j * 16 + l15;
        const v8f a = *accs[i][j];
#pragma unroll
        for (int r = 0; r < 8; ++r) {
          float dot = a[r];
          float out = (MODE == 1)
                          ? variance * __expf(sc * (zn[m + r] + pnv[x] - 2.f * dot))
                          : dot;
          D[(m + r) * XX + x] = out;
        }
      }
  }
}

// ---------------- column reductions ----------------
__global__ __launch_bounds__(256) void colreduce_kernel(const float* __restrict__ B1,
                                                        const float* __restrict__ Abuf,
                                                        const float* __restrict__ qmu, int g,
                                                        float* __restrict__ term1,
                                                        float* __restrict__ fmean) {
  int x = blockIdx.x * 256 + threadIdx.x;
  if (x >= XX) return;
  float t1 = 0.f, fm = 0.f;
  for (int m = 0; m < MM; ++m) {
    float a = Abuf[m * XX + x];
    t1 += B1[m * XX + x] * a;
    fm += qmu[m * GG + g] * a;
  }
  term1[x] = t1;
  fmean[x] = fm;
}

__global__ __launch_bounds__(256) void final_kernel(const float* __restrict__ fmean,
                                                    const float* __restrict__ term1,
                                                    const float* __restrict__ sqpart,
                                                    const float* __restrict__ varp, int g,
                                                    float* __restrict__ out) {
  int x = blockIdx.x * 256 + threadIdx.x;
  if (x >= XX) return;
  int p = x / NN, n = x - p * NN;
  float var = varp[0] - term1[x];
#pragma unroll
  for (int t = 0; t < MTILES; ++t) var += sqpart[t * XX + x];
  int o = n * (PPATCH * GG) + p * GG + g;
  out[o] = fmean[x];
  out[NN * PPATCH * GG + o] = var;
}

// ---------------- host ----------------
static size_t alignup(size_t v) { return (v + 255) & ~(size_t)255; }

extern "C" void kernel_launch(void* const* d_in, const int* in_sizes, int n_in,
                              void* d_out, int out_size, void* d_ws, size_t ws_size,
                              hipStream_t stream) {
  const float* ndx = (const float*)d_in[0];   // [32, 3072]
  const float* Z   = (const float*)d_in[1];   // [2, 384, 75]
  const float* qmu = (const float*)d_in[2];   // [384, 2]
  const float* qs  = (const float*)d_in[3];   // [2, 384, 384]
  const float* varP = (const float*)d_in[4];  // scalar
  const float* lsP  = (const float*)d_in[5];  // scalar
  float* out = (float*)d_out;

  char* ws = (char*)d_ws;
  size_t off = 0;
  auto take = [&](size_t bytes) { char* p = ws + off; off = alignup(off + bytes); return p; };
  float* patT  = (float*)take((size_t)LPAD * XX * 4);
  float* pn    = (float*)take((size_t)XX * 4);
  float* zn    = (float*)take((size_t)MM * 4);
  float* Zp    = (float*)take((size_t)MM * LPAD * 4);
  float* Kuu   = (float*)take((size_t)MM * MM * 4);
  float* W     = (float*)take((size_t)MM * MM * 4);
  float* S     = (float*)take((size_t)MM * MM * 4);
  float* LqT   = (float*)take((size_t)MM * MM * 4);
  float* B1    = (float*)take((size_t)MM * XX * 4);
  float* Abuf  = (float*)take((size_t)MM * XX * 4);
  float* term1 = (float*)take((size_t)XX * 4);
  float* fmean = (float*)take((size_t)XX * 4);
  float* sqpart = (float*)take((size_t)MTILES * XX * 4);
  (void)ws_size; (void)in_sizes; (void)n_in; (void)out_size;

  dim3 gemm_grid(XX / 128, MM / 64);  // (196, 6)

  patches_kernel<<<(LPAD * XX + 255) / 256, 256, 0, stream>>>(ndx, patT);
  pn_kernel<<<(XX + 255) / 256, 256, 0, stream>>>(patT, pn);

  for (int g = 0; g < GG; ++g) {
    const float* Zg = Z + (size_t)g * MM * LL;
    zn_kernel<<<(MM + 255) / 256, 256, 0, stream>>>(Zg, zn);
    zp_kernel<<<(MM * LPAD + 255) / 256, 256, 0, stream>>>(Zg, Zp);
    kuu_kernel<<<(MM * MM + 255) / 256, 256, 0, stream>>>(Zg, zn, varP, lsP, Kuu);
    chol_kernel<<<1, 256, 0, stream>>>(Kuu);
    winv_kernel<<<(MM + 255) / 256, 256, 0, stream>>>(Kuu, W);
    s_kernel<<<(MM * MM + 255) / 256, 256, 0, stream>>>(W, S);

    // B1 = RBF(Z, patches) : WMMA + RBF epilogue
    gemm_wmma<1><<<gemm_grid, 256, 0, stream>>>(Zp, patT, B1, LPAD, zn, pn, varP, lsP);
    // A = Kuu^-1 * B1 : WMMA
    gemm_wmma<0><<<gemm_grid, 256, 0, stream>>>(S, B1, Abuf, MM, zn, pn, varP, lsP);
    // term1 = diag(B1^T A), fmean = q_mu^T A
    colreduce_kernel<<<(XX + 255) / 256, 256, 0, stream>>>(B1, Abuf, qmu, g, term1, fmean);
    // sqpart = per-row-tile column sums of (Lq^T A)^2 : WMMA + fused reduction
    lqt_kernel<<<(MM * MM + 255) / 256, 256, 0, stream>>>(qs, g, LqT);
    gemm_wmma<2><<<gemm_grid, 256, 0, stream>>>(LqT, Abuf, sqpart, MM, zn, pn, varP, lsP);

    final_kernel<<<(XX + 255) / 256, 256, 0, stream>>>(fmean, term1, sqpart, varP, g, out);
  }
}